// GraphBlock_4492535791885
// MI455X (gfx1250) — compile-verified
//
#include <hip/hip_runtime.h>
#include <math.h>

typedef __attribute__((ext_vector_type(2))) float v2f;
typedef __attribute__((ext_vector_type(8))) float v8f;
typedef __attribute__((ext_vector_type(4))) unsigned int v4u;
typedef __attribute__((ext_vector_type(4))) int v4i;
typedef __attribute__((ext_vector_type(8))) int v8i;

#define D     128
#define KNN   16
#define ROWT  16
#define LDSP  132   // padded row stride (132 mod 64 = 4 -> conflict-free A-frag reads)

__device__ __forceinline__ float wave_reduce_sum(float v) {
#pragma unroll
    for (int m = 16; m >= 1; m >>= 1) v += __shfl_xor(v, m, 32);
    return v;
}

__device__ __forceinline__ float leaky(float a) { return a > 0.0f ? a : 0.01f * a; }

// ---------------------------------------------------------------------------
// Kernel 1: e_h = e_h_input @ Wh^T + bh ; e_t = e_t_input @ Wt^T + bt
// One wave per 16x16 output tile, fp32 WMMA, K-loop of 32 steps.
// ---------------------------------------------------------------------------
__global__ __launch_bounds__(256)
void embed_gemm_kernel(const float* __restrict__ inH, const float* __restrict__ inT,
                       const float* __restrict__ Wh,  const float* __restrict__ bh,
                       const float* __restrict__ Wt,  const float* __restrict__ bt,
                       float* __restrict__ outH, float* __restrict__ outT)
{
    const int wave = threadIdx.x >> 5;
    const int lane = threadIdx.x & 31;
    const int m16  = lane & 15;
    const int koff = (lane >> 4) << 1;          // 0 for lanes 0-15, 2 for lanes 16-31
    const int rowBase = blockIdx.x * ROWT;
    const int colBase = wave * 16;

    const bool  isT = (blockIdx.y != 0);
    const float* A  = isT ? inT  : inH;
    const float* W  = isT ? Wt   : Wh;
    const float* bv = isT ? bt   : bh;
    float*       out= isT ? outT : outH;

    const float bias = bv[colBase + m16];
    v8f acc;
#pragma unroll
    for (int r = 0; r < 8; ++r) acc[r] = bias;

    const float* Arow = A + (size_t)(rowBase + m16) * D;   // A fragment: row = M
    const float* Wrow = W + (size_t)(colBase + m16) * D;   // B fragment: B[k][n] = W[n*D+k]

#pragma unroll 4
    for (int kk = 0; kk < D; kk += 4) {
        v2f a, b;
        a.x = Arow[kk + koff];
        a.y = Arow[kk + koff + 1];
        b.x = Wrow[kk + koff];
        b.y = Wrow[kk + koff + 1];
        acc = __builtin_amdgcn_wmma_f32_16x16x4_f32(false, a, false, b,
                                                    (short)0, acc, false, false);
    }

#pragma unroll
    for (int r = 0; r < 8; ++r) {
        const int m = r + ((lane >> 4) << 3);
        out[(size_t)(rowBase + m) * D + colBase + m16] = acc[r];
    }
}

// ---------------------------------------------------------------------------
// Kernel 2: TDM gather-mode neighbor fetch -> gated attention + softmax +
// e_Nh, then fused (x+e_Nh)@W1^T and (x*e_Nh)@W2^T via WMMA, leaky-relu,
// layernorm.  16 nodes per block; 8 waves; 2 nodes per wave.
// Dynamic LDS: 8 waves * 2 slots * (16 rows * 128 f32) = 128 KB gather slab.
// ---------------------------------------------------------------------------
__global__ __launch_bounds__(256)
void node_fused_kernel(const float* __restrict__ x,
                       const float* __restrict__ e_h,
                       const float* __restrict__ e_t,
                       const int*   __restrict__ topk_idx,
                       const float* __restrict__ topk_w,
                       const float* __restrict__ W1, const float* __restrict__ b1,
                       const float* __restrict__ W2, const float* __restrict__ b2,
                       const float* __restrict__ gamma, const float* __restrict__ beta,
                       float* __restrict__ out, int N)
{
    __shared__ float lds_u[16][LDSP];
    __shared__ float lds_v[16][LDSP];
    __shared__ float lds_h[16][LDSP];
    extern __shared__ float gslab[];            // gather slots (dynamic LDS)

    const int wave = threadIdx.x >> 5;
    const int lane = threadIdx.x & 31;
    const int nodeBase = blockIdx.x * 16;

    // ---------- Issue TDM gather-mode loads: 16 neighbor rows per node -----
    const unsigned dynBase = __builtin_amdgcn_groupstaticsize();
#pragma unroll
    for (int s = 0; s < 2; ++s) {
        const int nl = wave * 2 + s;
        const int n  = nodeBase + nl;
        const int* idxPtr = topk_idx + (size_t)n * KNN;

        unsigned pk[8];                         // 16-bit row indices, packed
#pragma unroll
        for (int i = 0; i < 8; ++i) {
            const unsigned a = (unsigned)idxPtr[2 * i]     & 0xFFFFu;
            const unsigned b = (unsigned)idxPtr[2 * i + 1] & 0xFFFFu;
            pk[i] = (unsigned)__builtin_amdgcn_readfirstlane((int)(a | (b << 16)));
        }

        const unsigned ldsOff = (unsigned)__builtin_amdgcn_readfirstlane(
            (int)(dynBase + (unsigned)nl * (KNN * D * 4)));
        const unsigned long long ga = (unsigned long long)(uintptr_t)e_t;

        // D# group 0: count=1, gather_mode=1 (bit31), 16-bit idx (bit30=0),
        //             lds_addr, global_addr[56:0], type=2 (bits 127:126).
        v4u g0 = { 0x80000001u,
                   ldsOff,
                   (unsigned)(ga & 0xFFFFFFFFu),
                   ((unsigned)(ga >> 32) & 0x01FFFFFFu) | 0x80000000u };
        // D# group 1: data_size=4B (bits17:16=2); tensor_dim0=128;
        //             tensor_dim1=N; tile_dim0=128; tile_dim1=16 indices;
        //             tensor_dim0_stride=128.
        v8i g1 = { (int)0x00020000u,
                   (int)(128u << 16),                              // dim0[15:0]
                   (int)(((unsigned)N & 0xFFFFu) << 16),           // dim1[15:0]
                   (int)(((unsigned)N >> 16) | (128u << 16)),      // dim1 hi | tile_dim0
                   (int)KNN,                                       // tile_dim1
                   (int)D,                                         // dim0_stride lo
                   0, 0 };
        v4i g2 = { (int)pk[0], (int)pk[1], (int)pk[2], (int)pk[3] };
        v4i g3 = { (int)pk[4], (int)pk[5], (int)pk[6], (int)pk[7] };
        v8i g4 = { 0, 0, 0, 0, 0, 0, 0, 0 };    // unused trailing group (count=1)

        __builtin_amdgcn_tensor_load_to_lds(g0, g1, g2, g3, g4, 0);
    }
    __builtin_amdgcn_s_wait_tensorcnt(0);
    asm volatile("" ::: "memory");

    // ---------------- Phase A: per-node attention (2 nodes per wave) -------
    for (int s = 0; s < 2; ++s) {
        const int nl = wave * 2 + s;            // local node 0..15
        const int n  = nodeBase + nl;
        const int c0 = lane * 4;                // 4 features per lane
        const float* slot = gslab + (size_t)nl * (KNN * D);

        const float4 eh = *(const float4*)(e_h + (size_t)n * D + c0);
        const float4 xr = *(const float4*)(x   + (size_t)n * D + c0);

        float sc[KNN];
#pragma unroll
        for (int k = 0; k < KNN; ++k) {
            const float p = topk_w[n * KNN + k];
            const float q = 1.0f - p;
            const float4 nbv = *(const float4*)(slot + k * D + c0);
            float4 g;
            g.x = tanhf(eh.x + (p * nbv.x + q * eh.x));
            g.y = tanhf(eh.y + (p * nbv.y + q * eh.y));
            g.z = tanhf(eh.z + (p * nbv.z + q * eh.z));
            g.w = tanhf(eh.w + (p * nbv.w + q * eh.w));
            const float part = nbv.x * g.x + nbv.y * g.y + nbv.z * g.z + nbv.w * g.w;
            sc[k] = wave_reduce_sum(part);      // broadcast to all lanes
        }

        // softmax over K=16 (all lanes redundantly)
        float mx = sc[0];
#pragma unroll
        for (int k = 1; k < KNN; ++k) mx = fmaxf(mx, sc[k]);
        float pr[KNN];
        float ssum = 0.0f;
#pragma unroll
        for (int k = 0; k < KNN; ++k) { pr[k] = __expf(sc[k] - mx); ssum += pr[k]; }
        const float inv = 1.0f / ssum;

        float4 eN = make_float4(0.f, 0.f, 0.f, 0.f);
#pragma unroll
        for (int k = 0; k < KNN; ++k) {
            const float w = pr[k] * inv;
            const float4 nbv = *(const float4*)(slot + k * D + c0);
            eN.x += w * nbv.x; eN.y += w * nbv.y;
            eN.z += w * nbv.z; eN.w += w * nbv.w;
        }

        lds_u[nl][c0 + 0] = xr.x + eN.x;  lds_v[nl][c0 + 0] = xr.x * eN.x;
        lds_u[nl][c0 + 1] = xr.y + eN.y;  lds_v[nl][c0 + 1] = xr.y * eN.y;
        lds_u[nl][c0 + 2] = xr.z + eN.z;  lds_v[nl][c0 + 2] = xr.z * eN.z;
        lds_u[nl][c0 + 3] = xr.w + eN.w;  lds_v[nl][c0 + 3] = xr.w * eN.w;
    }
    __syncthreads();

    // ---------------- Phase B: WMMA GEMMs over W1 / W2 ---------------------
    {
        const int m16  = lane & 15;
        const int koff = (lane >> 4) << 1;
        const int colBase = wave * 16;

        const float bias1 = b1[colBase + m16];
        const float bias2 = b2[colBase + m16];
        v8f cu, cv;
#pragma unroll
        for (int r = 0; r < 8; ++r) { cu[r] = bias1; cv[r] = bias2; }

        const float* W1row = W1 + (size_t)(colBase + m16) * D;
        const float* W2row = W2 + (size_t)(colBase + m16) * D;

#pragma unroll 4
        for (int kk = 0; kk < D; kk += 4) {
            v2f au, av, bw1, bw2;
            au.x = lds_u[m16][kk + koff];  au.y = lds_u[m16][kk + koff + 1];
            av.x = lds_v[m16][kk + koff];  av.y = lds_v[m16][kk + koff + 1];
            bw1.x = W1row[kk + koff];      bw1.y = W1row[kk + koff + 1];
            bw2.x = W2row[kk + koff];      bw2.y = W2row[kk + koff + 1];
            cu = __builtin_amdgcn_wmma_f32_16x16x4_f32(false, au, false, bw1,
                                                       (short)0, cu, false, false);
            cv = __builtin_amdgcn_wmma_f32_16x16x4_f32(false, av, false, bw2,
                                                       (short)0, cv, false, false);
        }

#pragma unroll
        for (int r = 0; r < 8; ++r) {
            const int m = r + ((lane >> 4) << 3);
            lds_h[m][colBase + m16] = leaky(cu[r]) + leaky(cv[r]);
        }
    }
    __syncthreads();

    // ---------------- Phase C: layernorm (2 rows per wave) -----------------
    for (int s = 0; s < 2; ++s) {
        const int rl = wave * 2 + s;
        const int c0 = lane * 4;
        const float4 h = *(const float4*)(&lds_h[rl][c0]);

        const float sum = wave_reduce_sum(h.x + h.y + h.z + h.w);
        const float sq  = wave_reduce_sum(h.x * h.x + h.y * h.y + h.z * h.z + h.w * h.w);
        const float mean = sum * (1.0f / 128.0f);
        const float var  = sq * (1.0f / 128.0f) - mean * mean;
        const float rstd = rsqrtf(var + 1e-5f);

        const size_t o = (size_t)(nodeBase + rl) * D + c0;
        out[o + 0] = (h.x - mean) * rstd * gamma[c0 + 0] + beta[c0 + 0];
        out[o + 1] = (h.y - mean) * rstd * gamma[c0 + 1] + beta[c0 + 1];
        out[o + 2] = (h.z - mean) * rstd * gamma[c0 + 2] + beta[c0 + 2];
        out[o + 3] = (h.w - mean) * rstd * gamma[c0 + 3] + beta[c0 + 3];
    }
}

// ---------------------------------------------------------------------------
extern "C" void kernel_launch(void* const* d_in, const int* in_sizes, int n_in,
                              void* d_out, int out_size, void* d_ws, size_t ws_size,
                              hipStream_t stream)
{
    const float* x     = (const float*)d_in[0];
    const float* ehIn  = (const float*)d_in[1];
    const float* etIn  = (const float*)d_in[2];
    const int*   tIdx  = (const int*)  d_in[3];
    const float* tW    = (const float*)d_in[4];
    const float* Wh    = (const float*)d_in[5];
    const float* bh    = (const float*)d_in[6];
    const float* Wt    = (const float*)d_in[7];
    const float* bt    = (const float*)d_in[8];
    const float* W1    = (const float*)d_in[9];
    const float* b1    = (const float*)d_in[10];
    const float* W2    = (const float*)d_in[11];
    const float* b2    = (const float*)d_in[12];
    const float* gamma = (const float*)d_in[13];
    const float* beta  = (const float*)d_in[14];
    float* out = (float*)d_out;

    const int N = in_sizes[0] / D;              // 50000 (multiple of 16, < 65536)
    float* e_h = (float*)d_ws;
    float* e_t = e_h + (size_t)N * D;

    dim3 g1(N / ROWT, 2);
    embed_gemm_kernel<<<g1, 256, 0, stream>>>(ehIn, etIn, Wh, bh, Wt, bt, e_h, e_t);

    const size_t dynLds = 16u * KNN * D * sizeof(float);   // 128 KB gather slab
    node_fused_kernel<<<N / ROWT, 256, dynLds, stream>>>(x, e_h, e_t, tIdx, tW,
                                                         W1, b1, W2, b2, gamma, beta,
                                                         out, N);
}